// EikonalLoss_81518479278723
// MI455X (gfx1250) — compile-verified
//
#include <hip/hip_runtime.h>
#include <math.h>
#include <stdint.h>

#define H      512
#define W      512
#define TILE   64
#define HALO   2
#define BLOCK  256

#define WENO_EPS 1e-6f
#define EPS      1e-8f

typedef unsigned int u32v4 __attribute__((ext_vector_type(4)));
typedef int          i32v4 __attribute__((ext_vector_type(4)));
typedef int          i32v8 __attribute__((ext_vector_type(8)));
typedef float        f32x2 __attribute__((ext_vector_type(2)));

#if defined(__HIP_DEVICE_COMPILE__) && defined(__gfx1250__) && __has_builtin(__builtin_amdgcn_tensor_load_to_lds)
#define USE_TDM 1
#else
#define USE_TDM 0
#endif

__device__ __forceinline__ int imin(int a, int b) { return a < b ? a : b; }
__device__ __forceinline__ int imax(int a, int b) { return a > b ? a : b; }

__device__ __forceinline__ float frcp(float x) {
#if __has_builtin(__builtin_amdgcn_rcpf)
    return __builtin_amdgcn_rcpf(x);   // v_rcp_f32
#else
    return 1.0f / x;
#endif
}

__device__ __forceinline__ float fsqrt(float x) {
#if __has_builtin(__builtin_amdgcn_sqrtf)
    return __builtin_amdgcn_sqrtf(x);  // v_sqrt_f32
#else
    return sqrtf(x);
#endif
}

__device__ __forceinline__ f32x2 rcp2(f32x2 x) {
    f32x2 r; r.x = frcp(x.x); r.y = frcp(x.y); return r;
}

// t = (WENO_EPS + (a-b)^2)^2, packed
__device__ __forceinline__ f32x2 tsq2(f32x2 a, f32x2 b) {
    f32x2 s = a - b;
    f32x2 t = s * s + WENO_EPS;
    return t * t;
}

// One direction, both biased reconstructions, relu+max. All mul/add/fma packable.
__device__ __forceinline__ f32x2 weno_dir(f32x2 t01, f32x2 t12, f32x2 t23,
                                          f32x2 d0, f32x2 d1, f32x2 d2, f32x2 d3) {
    f32x2 n0 = (1.0f / 3.0f) * t12;                 // shared numerator
    f32x2 wm = n0 * rcp2(n0 + (2.0f / 3.0f) * t01); // left-biased weight
    f32x2 wp = n0 * rcp2(n0 + (2.0f / 3.0f) * t23); // right-biased weight
    f32x2 p1 = 0.5f * (d1 + d2);                    // shared central stencil
    f32x2 dm = d1 - 0.5f * (d0 + d2);               // p0m - p1
    f32x2 dp = d2 - 0.5f * (d3 + d1);               // p0p - p1
    f32x2 um = p1 + wm * dm;
    f32x2 up = p1 + wp * dp;
    f32x2 g;
    g.x = fmaxf(fmaxf(um.x, 0.0f), fmaxf(-up.x, 0.0f));
    g.y = fmaxf(fmaxf(um.y, 0.0f), fmaxf(-up.y, 0.0f));
    return g;
}

__device__ __forceinline__ void accum_px(float gmag, float rv, float target,
                                         float tLo, float tHi, float& num, float& den) {
    float resid = gmag - target;
    float ar    = fabsf(resid);
    float pp    = (ar < 0.01f) ? (50.0f * resid * resid) : (ar - 0.005f);
    float fm    = ((rv > 0.5f) && (gmag >= tLo) && (gmag <= tHi)) ? 1.0f : 0.0f;
    num += pp * fm;
    den += fm;
}

__global__ __launch_bounds__(BLOCK) void eik_tile_kernel(
    const float* __restrict__ pred, const float* __restrict__ reach,
    float2* __restrict__ partials, float target)
{
    __shared__ float smem[(TILE + 2 * HALO) * (TILE + 2 * HALO)];
    __shared__ float2 wpart[BLOCK / 32];

    const int tx0 = blockIdx.x * TILE;
    const int ty0 = blockIdx.y * TILE;
    const int b   = blockIdx.z;

    const int loadX0 = imax(tx0 - HALO, 0);
    const int loadX1 = imin(tx0 + TILE + HALO, W);
    const int loadY0 = imax(ty0 - HALO, 0);
    const int loadY1 = imin(ty0 + TILE + HALO, H);
    const int loadW  = loadX1 - loadX0;
    const int loadH  = loadY1 - loadY0;

    const float* predB  = pred  + (size_t)b * H * W;
    const float* reachB = reach + (size_t)b * H * W;
    const int tid = threadIdx.x;

#if USE_TDM
    if ((tid >> 5) == 0) {  // one full wave32 issues the TDM descriptor
        uint64_t gaddr = (uint64_t)(uintptr_t)(predB + (size_t)loadY0 * W + loadX0);
        uint32_t lds   = (uint32_t)(uintptr_t)(&smem[0]);
        u32v4 g0;
        g0[0] = 1u;                                                   // count=1
        g0[1] = lds;                                                  // lds_addr
        g0[2] = (uint32_t)gaddr;                                      // global_addr[31:0]
        g0[3] = (uint32_t)((gaddr >> 32) & 0x01FFFFFFu) | (2u << 30); // global_addr[56:32] | type=2
        i32v8 g1;
        g1[0] = (int)(2u << 16);                                      // data_size=4B
        g1[1] = (int)((uint32_t)W << 16);                             // tensor_dim0 lo
        g1[2] = (int)(((uint32_t)W >> 16) | ((uint32_t)H << 16));     // tensor_dim0 hi | tensor_dim1 lo
        g1[3] = (int)(((uint32_t)H >> 16) | ((uint32_t)loadW << 16)); // tensor_dim1 hi | tile_dim0
        g1[4] = (int)((uint32_t)loadH);                               // tile_dim1 | tile_dim2=0
        g1[5] = W;                                                    // tensor_dim0_stride lo
        g1[6] = 0;
        g1[7] = 0;
        i32v4 gz = {0, 0, 0, 0};
#if __clang_major__ >= 23
        i32v8 gz8 = {0, 0, 0, 0, 0, 0, 0, 0};
        __builtin_amdgcn_tensor_load_to_lds(g0, g1, gz, gz, gz8, 0);
#else
        __builtin_amdgcn_tensor_load_to_lds(g0, g1, gz, gz, 0);
#endif
#if __has_builtin(__builtin_amdgcn_s_wait_tensorcnt)
        __builtin_amdgcn_s_wait_tensorcnt(0);
#else
        asm volatile("s_wait_tensorcnt 0" ::: "memory");
#endif
    }
    __syncthreads();
#else
    for (int i = tid; i < loadW * loadH; i += BLOCK) {
        int r = i / loadW, c = i - r * loadW;
        smem[i] = predB[(size_t)(loadY0 + r) * W + (loadX0 + c)];
    }
    __syncthreads();
#endif

    // Each thread: fixed column; 16 rows processed as two 8-row streams packed in f32x2
    // (component .x = rows oyS..oyS+7, component .y = rows oyS+8..oyS+15).
    const int tx     = tid & (TILE - 1);
    const int ryBase = (tid >> 6) * 16;
    const int ox     = tx0 + tx;
    const int oyA    = ty0 + ryBase;
    const int oyB    = oyA + 8;
    const int base   = -loadY0;

    const int cx0 = imax(ox - 2, 0) - loadX0;
    const int cx1 = imax(ox - 1, 0) - loadX0;
    const int c2  = ox - loadX0;
    const int cx3 = imin(ox + 1, W - 1) - loadX0;
    const int cx4 = imin(ox + 2, W - 1) - loadX0;

#define SM(row, col) smem[((row) + base) * loadW + (col)]

    // init vertical windows for both streams
    f32x2 w0v, w1v, w2, w3, w4;
    w0v.x = SM(imax(oyA - 2, 0), c2);      w0v.y = SM(imax(oyB - 2, 0), c2);
    w1v.x = SM(imax(oyA - 1, 0), c2);      w1v.y = SM(imax(oyB - 1, 0), c2);
    w2.x  = SM(oyA, c2);                   w2.y  = SM(oyB, c2);
    w3.x  = SM(imin(oyA + 1, H - 1), c2);  w3.y  = SM(imin(oyB + 1, H - 1), c2);
    w4.x  = SM(imin(oyA + 2, H - 1), c2);  w4.y  = SM(imin(oyB + 2, H - 1), c2);

    f32x2 e0 = w1v - w0v, e1 = w2 - w1v, e2 = w3 - w2, e3 = w4 - w3;
    f32x2 yt01 = tsq2(e0, e1), yt12 = tsq2(e1, e2), yt23 = tsq2(e2, e3);

    const float tLo = 0.3f * target;
    const float tHi = 5.0f * target;

    float num = 0.0f, den = 0.0f;

#pragma unroll
    for (int r = 0; r < 8; ++r) {
        const int ryA = oyA + r;
        const int ryB = oyB + r;
        const float* prA = smem + (ryA + base) * loadW;
        const float* prB = smem + (ryB + base) * loadW;

        f32x2 a0, a1, a3, a4, wn, rv;
        a0.x = prA[cx0];  a0.y = prB[cx0];
        a1.x = prA[cx1];  a1.y = prB[cx1];
        a3.x = prA[cx3];  a3.y = prB[cx3];
        a4.x = prA[cx4];  a4.y = prB[cx4];
        const f32x2 u0 = w2;
        wn.x = SM(imin(ryA + 3, H - 1), c2);
        wn.y = SM(imin(ryB + 3, H - 1), c2);
        rv.x = reachB[(size_t)ryA * W + ox];
        rv.y = reachB[(size_t)ryB * W + ox];

        // x-direction
        f32x2 d0 = a1 - a0, d1 = u0 - a1, d2 = a3 - u0, d3 = a4 - a3;
        f32x2 xt01 = tsq2(d0, d1), xt12 = tsq2(d1, d2), xt23 = tsq2(d2, d3);
        f32x2 gx = weno_dir(xt01, xt12, xt23, d0, d1, d2, d3);

        // y-direction (rolled terms)
        f32x2 gy = weno_dir(yt01, yt12, yt23, e0, e1, e2, e3);

        f32x2 g2 = (gx * gx + gy * gy) + EPS;
        accum_px(fsqrt(g2.x), rv.x, target, tLo, tHi, num, den);
        accum_px(fsqrt(g2.y), rv.y, target, tLo, tHi, num, den);

        // roll window / diffs / smoothness terms
        w2 = w3; w3 = w4; w4 = wn;
        e0 = e1; e1 = e2; e2 = e3; e3 = w4 - w3;
        yt01 = yt12; yt12 = yt23; yt23 = tsq2(e2, e3);
    }
#undef SM

    // wave32 reduction
#pragma unroll
    for (int off = 16; off > 0; off >>= 1) {
        num += __shfl_down(num, off);
        den += __shfl_down(den, off);
    }
    if ((tid & 31) == 0) wpart[tid >> 5] = make_float2(num, den);
    __syncthreads();
    if (tid == 0) {
        float n = 0.0f, d = 0.0f;
#pragma unroll
        for (int i = 0; i < BLOCK / 32; ++i) { n += wpart[i].x; d += wpart[i].y; }
        const size_t bi = ((size_t)blockIdx.z * gridDim.y + blockIdx.y) * gridDim.x + blockIdx.x;
        partials[bi] = make_float2(n, d);
    }
}

__global__ __launch_bounds__(BLOCK) void eik_final_kernel(
    const float2* __restrict__ partials, int n, float* __restrict__ out)
{
    __shared__ float2 wpart[BLOCK / 32];
    float num = 0.0f, den = 0.0f;
    for (int i = threadIdx.x; i < n; i += BLOCK) {
        float2 p = partials[i];
        num += p.x;
        den += p.y;
    }
#pragma unroll
    for (int off = 16; off > 0; off >>= 1) {
        num += __shfl_down(num, off);
        den += __shfl_down(den, off);
    }
    if ((threadIdx.x & 31) == 0) wpart[threadIdx.x >> 5] = make_float2(num, den);
    __syncthreads();
    if (threadIdx.x == 0) {
        float n2 = 0.0f, d2 = 0.0f;
#pragma unroll
        for (int i = 0; i < BLOCK / 32; ++i) { n2 += wpart[i].x; d2 += wpart[i].y; }
        out[0] = n2 / fmaxf(d2, EPS);
    }
}

extern "C" void kernel_launch(void* const* d_in, const int* in_sizes, int n_in,
                              void* d_out, int out_size, void* d_ws, size_t ws_size,
                              hipStream_t stream) {
    const float* pred  = (const float*)d_in[0];
    const float* reach = (const float*)d_in[1];
    float* out = (float*)d_out;

    const int B = in_sizes[0] / (H * W);  // 32
    const int hh = H - 1, ww = W - 1;
    const float target = (float)(1.0 / sqrt((double)(hh * hh + ww * ww)));

    dim3 grid(W / TILE, H / TILE, B);
    float2* partials = (float2*)d_ws;
    const int nblocks = grid.x * grid.y * grid.z;

    eik_tile_kernel<<<grid, BLOCK, 0, stream>>>(pred, reach, partials, target);
    eik_final_kernel<<<1, BLOCK, 0, stream>>>(partials, nblocks, out);
}